// CTCLoss_10118942949576
// MI455X (gfx1250) — compile-verified
//
#include <hip/hip_runtime.h>
#include <hip/hip_bf16.h>
#include <cstdint>

#define BB 128
#define TT 160
#define CC 6625
#define LL 25
#define SS 51           // 2*L + 1
#define NEGV (-1e30f)

typedef float v2f __attribute__((ext_vector_type(2)));
typedef float v8f __attribute__((ext_vector_type(8)));

// ---------------------------------------------------------------------------
// Kernel 1: per-(b,t) logsumexp over C classes.
// Stream the 26.5 KB row global->LDS with CDNA5 async copy, two passes in LDS,
// then reduce the 256 exp-partials with v_wmma_f32_16x16x4_f32 (ones x data).
// ---------------------------------------------------------------------------
__global__ __launch_bounds__(256) void lse_kernel(const float* __restrict__ preds,
                                                  float* __restrict__ lse) {
    __shared__ float row[6628];   // 6625 floats padded to 16B multiple
    __shared__ float red[256];

    const int r   = blockIdx.x;            // r = b*T + t
    const int tid = threadIdx.x;
    const float* base = preds + (size_t)r * CC;

    // ---- async copy: 1656 x 16B chunks cover 6624 floats; 1 tail float ----
    for (int ch = tid; ch < 1656; ch += 256) {
        uint32_t lds_addr = (uint32_t)(uintptr_t)(&row[ch * 4]); // low 32 bits = LDS offset
        const float* g = base + ch * 4;
        asm volatile("global_load_async_to_lds_b128 %0, %1, off"
                     :: "v"(lds_addr), "v"(g) : "memory");
    }
    if (tid == 0) row[6624] = base[6624];
    asm volatile("s_wait_asynccnt 0" ::: "memory");
    __syncthreads();

    // ---- pass 1: row max ----
    float m = -3.402823466e+38f;
    for (int i = tid; i < CC; i += 256) m = fmaxf(m, row[i]);
    red[tid] = m;
    __syncthreads();
    for (int off = 128; off > 0; off >>= 1) {
        if (tid < off) red[tid] = fmaxf(red[tid], red[tid + off]);
        __syncthreads();
    }
    m = red[0];
    __syncthreads();

    // ---- pass 2: sum of exp(x - m) ----
    float s = 0.0f;
    for (int i = tid; i < CC; i += 256) s += expf(row[i] - m);
    red[tid] = s;
    __syncthreads();

    // ---- WMMA reduction of 256 partials by wave 0 ----
    // D = ones(16x4) * B(4x16) + C  ==> every row of D holds the column sums,
    // so accumulating 4 tiles and summing one 16-wide row gives the total.
    if (tid < 32) {
        v2f ones; ones.x = 1.0f; ones.y = 1.0f;
        v8f acc = {0.f, 0.f, 0.f, 0.f, 0.f, 0.f, 0.f, 0.f};
#pragma unroll
        for (int g = 0; g < 4; ++g) {
            v2f bmat;
            bmat.x = red[g * 64 + tid];
            bmat.y = red[g * 64 + 32 + tid];
            acc = __builtin_amdgcn_wmma_f32_16x16x4_f32(
                false, ones, false, bmat, (short)0, acc, false, false);
        }
        red[tid] = acc[0];   // lanes 0..15 of VGPR0 = row M=0 (N = lane)
    }
    if (tid == 0) {
        float total = 0.0f;
#pragma unroll
        for (int i = 0; i < 16; ++i) total += red[i];
        lse[r] = m + logf(total);
    }
}

// ---------------------------------------------------------------------------
// Kernel 2: CTC forward DP per batch (replicates the lax.scan exactly).
// ---------------------------------------------------------------------------
__device__ __forceinline__ float lae(float x, float y) {
    float m = fmaxf(x, y);
    float d = fabsf(x - y);
    return m + log1pf(expf(-d));
}

__global__ __launch_bounds__(64) void ctc_dp_kernel(const float* __restrict__ preds,
                                                    const int* __restrict__ labels,
                                                    const int* __restrict__ lens,
                                                    const float* __restrict__ lse,
                                                    float* __restrict__ loss_out) {
    const int b = blockIdx.x;
    const int s = threadIdx.x;

    __shared__ int   ext[SS];
    __shared__ int   allow[SS];
    __shared__ float bufA[SS], bufB[SS];

    if (s < SS) ext[s] = (s & 1) ? labels[b * LL + (s >> 1)] : 0;
    __syncthreads();
    if (s < SS) {
        int e = ext[s];
        int em2 = (s >= 2) ? ext[s - 2] : 0;
        allow[s] = (e != 0 && e != em2 && s >= 2) ? 1 : 0;
    }

    const float* prow = preds + (size_t)b * TT * CC;
    const float* lrow = lse + b * TT;

    // t = 0
    if (s < SS) {
        float lp = prow[ext[s]] - lrow[0];
        bufA[s] = (s < 2) ? lp : NEGV;
    }
    __syncthreads();

    float* cur = bufA;
    float* nxt = bufB;
    for (int t = 1; t < TT; ++t) {
        if (s < SS) {
            float x = cur[s];
            float y = (s >= 1) ? cur[s - 1] : NEGV;
            float acc = lae(x, y);
            if (allow[s]) acc = lae(acc, cur[s - 2]);
            float lp = prow[(size_t)t * CC + ext[s]] - lrow[t];
            nxt[s] = acc + lp;
        }
        __syncthreads();
        float* tmp = cur; cur = nxt; nxt = tmp;
    }

    if (s == 0) {
        int sl = 2 * lens[b];
        float loss = -lae(cur[sl], cur[sl - 1]);
        float w = 1.0f - expf(-loss);      // focal weight
        loss_out[b] = loss * (w * w);
    }
}

// ---------------------------------------------------------------------------
// Kernel 3: mean over the 128 per-batch losses.
// ---------------------------------------------------------------------------
__global__ __launch_bounds__(128) void mean_kernel(const float* __restrict__ loss,
                                                   float* __restrict__ out) {
    __shared__ float red[128];
    int i = threadIdx.x;
    red[i] = loss[i];
    __syncthreads();
    for (int off = 64; off > 0; off >>= 1) {
        if (i < off) red[i] += red[i + off];
        __syncthreads();
    }
    if (i == 0) out[0] = red[0] * (1.0f / (float)BB);
}

// ---------------------------------------------------------------------------
extern "C" void kernel_launch(void* const* d_in, const int* in_sizes, int n_in,
                              void* d_out, int out_size, void* d_ws, size_t ws_size,
                              hipStream_t stream) {
    const float* predicts = (const float*)d_in[0];   // [B, T, C] f32
    const int*   labels   = (const int*)d_in[1];     // [B, L] i32
    const int*   lens     = (const int*)d_in[2];     // [B] i32
    float* out = (float*)d_out;                      // scalar f32

    float* lse  = (float*)d_ws;                      // B*T floats = 80 KB
    float* loss = lse + BB * TT;                     // 128 floats

    lse_kernel<<<BB * TT, 256, 0, stream>>>(predicts, lse);
    ctc_dp_kernel<<<BB, 64, 0, stream>>>(predicts, labels, lens, lse, loss);
    mean_kernel<<<1, 128, 0, stream>>>(loss, out);
}